// Bilinear_70222715290053
// MI455X (gfx1250) — compile-verified
//
#include <hip/hip_runtime.h>

// Problem constants from the reference.
#define IMG_H 224
#define IMG_W 224
#define IMG_B 128
#define NCH_IN 5
#define NCH_OUT 3

// Memory-bound gather:
//   ~128.5 MB input read + ~77 MB output write => ~205 MB @ 23.3 TB/s ~= 9 us.
// Grid maps directly onto tensor geometry to eliminate all integer division:
//   block.x = 224 threads  (one image row, 7 full wave32s, EXEC all ones)
//   grid    = (H, B)       (row index, batch index)
// Data movement:
//  - dx/dy: one merged global_load_b64 per lane (stride-20B records; the RGB in
//    those same cachelines is consumed by the near-identity gather via L2)
//  - gather: merged b96 load; dx,dy ~ N(0,1) so targets are near-identity and
//    hit lines already resident in the 192 MB L2 (input is 128.5 MB total)
//  - output: non-temporal merged b96 store; streamed once, never re-read, so
//    it does not evict the L2-resident input
//  - floor-mod = x - floor(x/D)*D with IEEE division: bit-matches jnp.mod;
//    indices clamped to [0, D-1] mirroring JAX's index clamping on the
//    rounding edge case (m rounding up to exactly D).
__global__ __launch_bounds__(IMG_W) void
gather_warp_kernel(const float* __restrict__ x, float* __restrict__ out) {
    const int c  = threadIdx.x;            // column 0..223
    const int r  = blockIdx.x;             // row    0..223
    const int b  = blockIdx.y;             // batch  0..127
    const int bH = b * IMG_H;              // scalar (SALU)
    const int rc = bH + r;                 // b*H + r (scalar)
    const int p  = rc * IMG_W + c;         // pixel id (one v_add per lane)

    const float* rec = x + (size_t)p * NCH_IN;
    const float dx = rec[3];
    const float dy = rec[4];

    const float xv = (float)c + dx;
    const float yv = (float)r + dy;

    // jnp.mod(v, D) = v - floor(v/D)*D  (result sign follows divisor > 0)
    const float xm = xv - floorf(xv / (float)IMG_W) * (float)IMG_W;
    const float ym = yv - floorf(yv / (float)IMG_H) * (float)IMG_H;

    int Xi = (int)xm;
    int Yi = (int)ym;
    Xi = Xi < 0 ? 0 : (Xi > IMG_W - 1 ? IMG_W - 1 : Xi);
    Yi = Yi < 0 ? 0 : (Yi > IMG_H - 1 ? IMG_H - 1 : Yi);

    // gather RGB from (b, Yi, Xi)
    const float* src = x + ((size_t)(bH + Yi) * IMG_W + (size_t)Xi) * NCH_IN;
    const float g0 = src[0];
    const float g1 = src[1];
    const float g2 = src[2];

    // stream output with non-temporal hint
    float* o = out + (size_t)p * NCH_OUT;
    __builtin_nontemporal_store(g0, o + 0);
    __builtin_nontemporal_store(g1, o + 1);
    __builtin_nontemporal_store(g2, o + 2);
}

extern "C" void kernel_launch(void* const* d_in, const int* in_sizes, int n_in,
                              void* d_out, int out_size, void* d_ws, size_t ws_size,
                              hipStream_t stream) {
    const float* x = (const float*)d_in[0];  // (128,224,224,5) f32
    float* out = (float*)d_out;              // (128,224,224,3) f32
    (void)in_sizes; (void)n_in; (void)out_size; (void)d_ws; (void)ws_size;

    dim3 grid(IMG_H, IMG_B, 1);              // 224 x 128 blocks
    dim3 block(IMG_W, 1, 1);                 // 224 threads = 7 wave32s
    gather_warp_kernel<<<grid, block, 0, stream>>>(x, out);
}